// ViterbiDecoder_15857019257063
// MI455X (gfx1250) — compile-verified
//
#include <hip/hip_runtime.h>
#include <stdint.h>

// Viterbi decoder for MI455X (gfx1250).
// B=32 blocks (one per batch chain), 256 threads (8 wave32 waves) each.
// CDNA5 path: async global->LDS DMA ring buffer (DEPTH=8 stages, 16KB each)
// with ASYNCcnt-based software pipelining to hide HBM latency on the serial
// 512-step max-plus recurrence. Backpointers live in LDS (u8), backtrace done
// in-workgroup.

#define B_   32
#define S_   512
#define T_   64
#define DEPTH 8                       // ring-buffer stages (power of two)
#define TILE_FLOATS (T_ * T_)         // 4096 floats
#define TILE_BYTES  (TILE_FLOATS * 4) // 16 KB
// per wave, each stage issues 4 async b128 instructions (4 per thread)
// steady-state wait threshold: (DEPTH-1)*4 = 28

// Exact parameter types for the async-LDS builtin:
//   arg0: int32x4 in global (AS1), arg1: int32x4 in LDS (AS3), then imm,imm.
typedef int v4i __attribute__((vector_size(16)));
typedef __attribute__((address_space(1))) v4i v4i_g;
typedef __attribute__((address_space(3))) v4i v4i_l;

#if __has_builtin(__builtin_amdgcn_global_load_async_to_lds_b128)
__device__ __forceinline__ void async_cp16(const float* g, float* l) {
  __builtin_amdgcn_global_load_async_to_lds_b128(
      (v4i_g*)(uintptr_t)g,
      (v4i_l*)(uint32_t)(uintptr_t)l,
      0, 0);
}
#else
__device__ __forceinline__ void async_cp16(const float* g, float* l) {
  uint32_t loff = (uint32_t)(uintptr_t)l;       // LDS byte address (addr[31:0])
  uint64_t ga   = (uint64_t)(uintptr_t)g;       // 64-bit global address
  asm volatile("global_load_async_to_lds_b128 %0, %1, off offset:0"
               :: "v"(loff), "v"(ga) : "memory");
}
#endif

#if __has_builtin(__builtin_amdgcn_s_wait_asynccnt)
#define WAIT_ASYNC(n) __builtin_amdgcn_s_wait_asynccnt(n)
#else
#define WAIT_ASYNC(n) asm volatile("s_wait_asynccnt " #n ::: "memory")
#endif

__global__ void __launch_bounds__(256)
viterbi_kernel(const float* __restrict__ lp,
               const uint8_t* __restrict__ mask,
               int* __restrict__ out) {
  extern __shared__ char smem[];
  float*   buf   = (float*)smem;                              // DEPTH * 4096 f32
  float*   alpha = (float*)(smem + DEPTH * TILE_BYTES);       // 64 f32
  float*   pval  = alpha + T_;                                // 4*64 f32 partial max
  int*     pidx  = (int*)(pval + 4 * T_);                     // 4*64 i32 partial argmax
  uint8_t* bp    = (uint8_t*)(pidx + 4 * T_);                 // (S-1)*64 backpointers

  const int tid = (int)threadIdx.x;
  const int b   = (int)blockIdx.x;
  const int j   = tid & (T_ - 1);   // column (new tag)
  const int g   = tid >> 6;         // i-group 0..3 (16 rows each)

  const float* base = lp + (size_t)b * S_ * TILE_FLOATS;

  // Issue one full stage's async copies: 256 threads x 4 x 16B = 16 KB.
  auto issue = [&](int stage_src, int slot) {
    const float* gs = base + (size_t)stage_src * TILE_FLOATS;
    float*       ls = buf + slot * TILE_FLOATS;
    const int o = tid * 4;
    async_cp16(gs + o,        ls + o);
    async_cp16(gs + o + 1024, ls + o + 1024);
    async_cp16(gs + o + 2048, ls + o + 2048);
    async_cp16(gs + o + 3072, ls + o + 3072);
  };

  // Prologue: preload stages 0 .. DEPTH-2.
  for (int t = 0; t < DEPTH - 1; ++t) issue(t, t);

  for (int s = 0; s < S_; ++s) {
    // Keep exactly 4 async instrs issued per wave per iteration (dummy reload
    // of the current stage into the already-consumed slot near the tail) so
    // the in-order ASYNCcnt wait threshold stays uniform.
    const int nxt = s + DEPTH - 1;
    issue(nxt < S_ ? nxt : s, nxt & (DEPTH - 1));

    WAIT_ASYNC(28);        // stage s's 4 oldest loads (this wave) complete
    __syncthreads();       // all waves' portions of stage s visible in LDS

    const float* p = buf + (s & (DEPTH - 1)) * TILE_FLOATS;

    float best = -__builtin_inff();
    int   bi   = g * 16;
    if (s == 0) {
      // Free initial previous tag: alpha0[j] = max_i P[0][i][j].
#pragma unroll
      for (int k = 0; k < 16; ++k) {
        const int i = g * 16 + k;
        const float v = p[i * T_ + j];
        if (v > best) { best = v; bi = i; }
      }
    } else {
#pragma unroll
      for (int k = 0; k < 16; ++k) {
        const int i = g * 16 + k;
        const float v = alpha[i] + p[i * T_ + j];
        if (v > best) { best = v; bi = i; }
      }
    }
    pval[g * T_ + j] = best;
    pidx[g * T_ + j] = bi;
    __syncthreads();

    if (tid < T_) {
      // Combine groups in ascending-i order with strict '>' to preserve
      // jnp.argmax first-occurrence semantics.
      float bv = pval[j];
      int   ii = pidx[j];
#pragma unroll
      for (int g2 = 1; g2 < 4; ++g2) {
        const float v = pval[g2 * T_ + j];
        if (v > bv) { bv = v; ii = pidx[g2 * T_ + j]; }
      }
      alpha[j] = bv;
      if (s > 0) bp[(s - 1) * T_ + j] = (uint8_t)ii;
    }
    __syncthreads();
  }

  WAIT_ASYNC(0);           // drain tail dummy loads
  __syncthreads();

  if (tid == 0) {
    // last_tag = argmax_j alpha_last[j] (first occurrence).
    float bv = alpha[0];
    int tag = 0;
    for (int j2 = 1; j2 < T_; ++j2)
      if (alpha[j2] > bv) { bv = alpha[j2]; tag = j2; }
    // Backtrace and emit int32 tags, applying mask -> PADDING_INDEX.
    for (int s2 = S_ - 1; s2 >= 0; --s2) {
      out[b * S_ + s2] = mask[b * S_ + s2] ? tag : -1;
      if (s2 > 0) tag = (int)bp[(s2 - 1) * T_ + tag];
    }
  }
}

extern "C" void kernel_launch(void* const* d_in, const int* in_sizes, int n_in,
                              void* d_out, int out_size, void* d_ws, size_t ws_size,
                              hipStream_t stream) {
  (void)in_sizes; (void)n_in; (void)out_size; (void)d_ws; (void)ws_size;
  const float*   lp   = (const float*)d_in[0];
  const uint8_t* mask = (const uint8_t*)d_in[1];
  int*           out  = (int*)d_out;

  const size_t shmem = (size_t)DEPTH * TILE_BYTES    // tile ring buffer (128 KB)
                     + (size_t)T_ * 4                // alpha
                     + (size_t)4 * T_ * 4            // pval
                     + (size_t)4 * T_ * 4            // pidx
                     + (size_t)(S_ - 1) * T_;        // backpointers (u8)

  // Allow >64KB dynamic LDS (gfx1250 WGP has 320KB).
  (void)hipFuncSetAttribute((const void*)viterbi_kernel,
                            hipFuncAttributeMaxDynamicSharedMemorySize,
                            (int)shmem);

  viterbi_kernel<<<dim3(B_), dim3(256), shmem, stream>>>(lp, mask, out);
}